// MultiBoxLoss_30485677867282
// MI455X (gfx1250) — compile-verified
//
#include <hip/hip_runtime.h>
#include <math.h>

#define BATCH 8
#define NPRI  57744
#define NCLS  81
#define NOBJ  20
#define TOT   (BATCH*NPRI)
#define KEY_NEGINF 0x007FFFFFu     // okey(-inf); all finite keys are >= 0x00800000

typedef __attribute__((ext_vector_type(2))) float v2f;
typedef __attribute__((ext_vector_type(8))) float v8f;

// ---- workspace layout (bytes) ----
#define WS_ACC  0                        // 16 u32 accumulator words
#define WS_GT   256                      // u64[BATCH*NOBJ] packed per-GT best prior
#define WS_H1   2048                     // u32[4096]  bits[31:20]
#define WS_H2   (WS_H1 + 4096*4)         // u32[4096]  bits[19:8]
#define WS_H3   (WS_H2 + 4096*4)         // u32[256]   bits[7:0]
#define WS_BTO  36864                    // f32[TOT] best_truth_overlap
#define WS_BTI  (WS_BTO + TOT*4)         // i32[TOT] best_truth_idx
#define WS_CT   (WS_BTI + TOT*4)         // i32[TOT] conf_t
#define WS_KEY  (WS_CT  + TOT*4)         // u32[TOT] ordered score keys
#define WS_LIST (WS_KEY + TOT*4)         // u32[TOT] compacted kept-row ids
// acc slots: 0 sl1(f32) 1 num_pos(i32) 2 ce(f32) 3 keep_cnt(i32) 4 ticket(i32)
//            5 thr(u32) 6 rEq(i32) 7 p1(u32) 8 ca(u32) 9 p2(u32) 10 cb(u32) 11 done(u32)

__device__ __forceinline__ unsigned okey(float f) {
  unsigned u = __float_as_uint(f);
  return (u & 0x80000000u) ? ~u : (u | 0x80000000u);
}
__device__ __forceinline__ float sl1f(float d) {
  float a = fabsf(d);
  return a < 1.f ? 0.5f * d * d : a - 0.5f;
}

__global__ void k_zero(unsigned* w, int n) {
  int i = blockIdx.x * blockDim.x + threadIdx.x;
  if (i < n) w[i] = 0u;
}

// ---- K1: per-prior best GT + per-GT best prior (filtered packed atomicMax) ----
__global__ void k_match(const float4* __restrict__ priors,
                        const float4* __restrict__ gt, char* ws) {
  __shared__ float4 sg[NOBJ];
  __shared__ unsigned long long sb[NOBJ];
  const int b = blockIdx.y, t = threadIdx.x;
  if (t < NOBJ) { sg[t] = gt[b * NOBJ + t]; sb[t] = 0ull; }
  __syncthreads();
  const int p = blockIdx.x * blockDim.x + t;
  float* bto = (float*)(ws + WS_BTO);
  int*   bti = (int*)(ws + WS_BTI);
  if (p < NPRI) {
    float4 pr = priors[p];
    float px1 = pr.x - 0.5f * pr.z, py1 = pr.y - 0.5f * pr.w;
    float px2 = pr.x + 0.5f * pr.z, py2 = pr.y + 0.5f * pr.w;
    float ab = pr.z * pr.w;
    float biou = 0.f; int bidx = 0;
    #pragma unroll
    for (int n = 0; n < NOBJ; ++n) {
      float4 g = sg[n];
      float iw = fmaxf(fminf(g.z, px2) - fmaxf(g.x, px1), 0.f);
      float ih = fmaxf(fminf(g.w, py2) - fmaxf(g.y, py1), 0.f);
      float inter = iw * ih;
      float aa = (g.z - g.x) * (g.w - g.y);
      float iou = inter / (aa + ab - inter);
      if (iou > biou) { biou = iou; bidx = n; }
      unsigned long long key =
          ((unsigned long long)__float_as_uint(iou) << 32) |
          (unsigned long long)(0xFFFFFFFFu - (unsigned)p);
      if (key > sb[n]) atomicMax(&sb[n], key);   // monotonic filter: stale-read safe
    }
    bto[b * NPRI + p] = biou;
    bti[b * NPRI + p] = bidx;
  }
  __syncthreads();
  if (t < NOBJ) {
    unsigned long long* gb = (unsigned long long*)(ws + WS_GT) + b * NOBJ + t;
    if (sb[t] > *gb) atomicMax(gb, sb[t]);
  }
}

// ---- K2: force-match each GT to its best prior ----
__global__ void k_force(char* ws) {
  int b = threadIdx.x;
  if (b >= BATCH) return;
  unsigned long long* gb = (unsigned long long*)(ws + WS_GT);
  float* bto = (float*)(ws + WS_BTO);
  int*   bti = (int*)(ws + WS_BTI);
  for (int n = 0; n < NOBJ; ++n) {
    unsigned p = 0xFFFFFFFFu - (unsigned)(gb[b * NOBJ + n] & 0xFFFFFFFFull);
    bto[b * NPRI + p] = 2.0f;
    bti[b * NPRI + p] = n;
  }
}

// ---- K3: conf_t, encode, smooth-L1; chained WMMA reduction (no DS ops) ----
__global__ void k_loc(const float4* __restrict__ loc,
                      const float4* __restrict__ priors,
                      const float4* __restrict__ gt,
                      const int* __restrict__ glab, char* ws) {
  const float* bto = (const float*)(ws + WS_BTO);
  const int*   bti = (const int*)(ws + WS_BTI);
  int* ct = (int*)(ws + WS_CT);
  float s[2] = {0.f, 0.f};
  bool pos[2] = {false, false};
  int base = (blockIdx.x * blockDim.x + threadIdx.x) * 2;
  #pragma unroll
  for (int i = 0; i < 2; ++i) {
    int idx = base + i;
    if (idx < TOT) {
      int b = idx / NPRI;
      int p = idx - b * NPRI;
      float ov = bto[idx];
      int ti = bti[idx];
      int c = ov < 0.4f ? 0 : (ov < 0.5f ? -1 : glab[b * NOBJ + ti]);
      ct[idx] = c;
      if (c > 0) {
        float4 g = gt[b * NOBJ + ti];
        float4 pr = priors[p];
        float t0 = ((g.x + g.z) * 0.5f - pr.x) / (0.1f * pr.z);
        float t1 = ((g.y + g.w) * 0.5f - pr.y) / (0.1f * pr.w);
        float t2 = logf((g.z - g.x) / pr.z) * 5.0f;
        float t3 = logf((g.w - g.y) / pr.w) * 5.0f;
        float4 ld = loc[idx];
        s[i] = sl1f(ld.x - t0) + sl1f(ld.y - t1) + sl1f(ld.z - t2) + sl1f(ld.w - t3);
        pos[i] = true;
      }
    }
  }
  // positives: ballot + popcount (VALU/SALU only, uniform across wave)
  int pc = __popcll(__ballot(pos[0])) + __popcll(__ballot(pos[1]));
  // 64 partials -> grand total with two chained V_WMMA_F32_16X16X4_F32
  v2f a;  a.x = s[0]; a.y = s[1];
  v2f bo; bo.x = 1.f; bo.y = 1.f;
  v8f c0 = {};
  c0 = __builtin_amdgcn_wmma_f32_16x16x4_f32(false, a, false, bo, (short)0, c0, false, false);
  v2f a2; a2.x = c0[0]+c0[1]+c0[2]+c0[3]+c0[4]+c0[5]+c0[6]+c0[7]; a2.y = 0.f;
  v8f c1 = {};
  c1 = __builtin_amdgcn_wmma_f32_16x16x4_f32(false, a2, false, bo, (short)0, c1, false, false);
  if ((threadIdx.x & 31) == 0) {
    atomicAdd((float*)(ws + WS_ACC) + 0, c1[0]);
    atomicAdd((int*)(ws + WS_ACC) + 1, pc);
  }
}

// ---- K4: mining score; 4 lanes per row, 8 rows per wave (0.25 bpermute/row) ----
__global__ void k_score(const float* __restrict__ conf, char* ws) {
  const int wave = blockIdx.x * (blockDim.x >> 5) + (threadIdx.x >> 5);
  const int lane = threadIdx.x & 31;
  const int g = lane >> 2, l = lane & 3;
  const int row = wave * 8 + g;
  size_t base = (size_t)row * NCLS;
  float m = (l > 0) ? conf[base + l] : -INFINITY;   // exclude background class 0
  #pragma unroll
  for (int i = 1; i < 20; ++i) m = fmaxf(m, conf[base + l + 4 * i]);
  if (l == 0) m = fmaxf(m, conf[base + 80]);
  m = fmaxf(m, __shfl_xor(m, 1, 32));
  m = fmaxf(m, __shfl_xor(m, 2, 32));
  if (l == 0) {
    bool cand = ((const int*)(ws + WS_CT))[row] == 0;
    ((unsigned*)(ws + WS_KEY))[row] = okey(cand ? m : -INFINITY);
  }
}

// ---- K5: 3-level radix select with LDS-privatized histograms ----
__global__ void k_histA(char* ws) {
  __shared__ unsigned h[4096];
  for (int j = threadIdx.x; j < 4096; j += blockDim.x) h[j] = 0u;
  __syncthreads();
  int i = blockIdx.x * blockDim.x + threadIdx.x;
  if (i < TOT) {
    unsigned key = ((const unsigned*)(ws + WS_KEY))[i];
    if (key != KEY_NEGINF) atomicAdd(&h[key >> 20], 1u);
  }
  __syncthreads();
  for (int j = threadIdx.x; j < 4096; j += blockDim.x) {
    unsigned c = h[j];
    if (c) atomicAdd((unsigned*)(ws + WS_H1) + j, c);
  }
}
__global__ void k_scanA(char* ws) {
  unsigned* acc = (unsigned*)(ws + WS_ACC);
  unsigned* h = (unsigned*)(ws + WS_H1);
  unsigned K = 3u * (unsigned)((int*)acc)[1];
  unsigned p1 = 0xFFFFFFFFu, ca = 0, done = 1, thr = 0xFFFFFFFFu; int rEq = 0;
  if (K > 0) {
    unsigned cum = 0; int found = 0;
    for (int bin = 4095; bin >= 0; --bin) {
      unsigned c = h[bin];
      if (cum + c >= K) { p1 = (unsigned)bin; ca = cum; found = 1; done = 0; break; }
      cum += c;
    }
    if (!found) { thr = 0u; rEq = 0x3FFFFFFF; }   // K > #candidates: accept all
  }
  acc[5] = thr; ((int*)acc)[6] = rEq; acc[7] = p1; acc[8] = ca; acc[11] = done;
}
__global__ void k_histB(char* ws) {
  __shared__ unsigned h[4096];
  for (int j = threadIdx.x; j < 4096; j += blockDim.x) h[j] = 0u;
  __syncthreads();
  unsigned p1 = ((const unsigned*)(ws + WS_ACC))[7];
  int i = blockIdx.x * blockDim.x + threadIdx.x;
  if (i < TOT) {
    unsigned key = ((const unsigned*)(ws + WS_KEY))[i];
    if ((key >> 20) == p1) atomicAdd(&h[(key >> 8) & 0xFFFu], 1u);
  }
  __syncthreads();
  for (int j = threadIdx.x; j < 4096; j += blockDim.x) {
    unsigned c = h[j];
    if (c) atomicAdd((unsigned*)(ws + WS_H2) + j, c);
  }
}
__global__ void k_scanB(char* ws) {
  unsigned* acc = (unsigned*)(ws + WS_ACC);
  if (acc[11]) { acc[9] = 0xFFFFFFFFu; return; }
  unsigned* h = (unsigned*)(ws + WS_H2);
  unsigned K = 3u * (unsigned)((int*)acc)[1];
  unsigned p1 = acc[7], ca = acc[8];
  unsigned Kr = K - ca, cum = 0, p2 = 0xFFFFFFFFu, cb = 0;
  int found = 0;
  for (int bin = 4095; bin >= 0; --bin) {
    unsigned c = h[bin];
    if (cum + c >= Kr) { p2 = (p1 << 12) | (unsigned)bin; cb = ca + cum; found = 1; break; }
    cum += c;
  }
  if (!found) { acc[5] = p1 << 20; ((int*)acc)[6] = 0x3FFFFFFF; acc[11] = 1; }
  acc[9] = p2; acc[10] = cb;
}
__global__ void k_histC(char* ws) {
  __shared__ unsigned h[256];
  if (threadIdx.x < 256) h[threadIdx.x] = 0u;
  __syncthreads();
  unsigned p2 = ((const unsigned*)(ws + WS_ACC))[9];
  int i = blockIdx.x * blockDim.x + threadIdx.x;
  if (i < TOT) {
    unsigned key = ((const unsigned*)(ws + WS_KEY))[i];
    if ((key >> 8) == p2) atomicAdd(&h[key & 0xFFu], 1u);
  }
  __syncthreads();
  if (threadIdx.x < 256) {
    unsigned c = h[threadIdx.x];
    if (c) atomicAdd((unsigned*)(ws + WS_H3) + threadIdx.x, c);
  }
}
__global__ void k_scanC(char* ws) {
  unsigned* acc = (unsigned*)(ws + WS_ACC);
  if (acc[11]) return;
  unsigned* h = (unsigned*)(ws + WS_H3);
  unsigned K = 3u * (unsigned)((int*)acc)[1];
  unsigned p2 = acc[9], cb = acc[10];
  unsigned Kr = K - cb, cum = 0;
  unsigned thr = p2 << 8; int rEq = 0x3FFFFFFF;
  for (int bin = 255; bin >= 0; --bin) {
    unsigned c = h[bin];
    if (cum + c >= Kr) { thr = (p2 << 8) | (unsigned)bin; rEq = (int)(Kr - cum); break; }
    cum += c;
  }
  acc[5] = thr; ((int*)acc)[6] = rEq;
}

// ---- K6a: compact kept rows (positives + hard negatives) into a list ----
__global__ void k_select(char* ws) {
  int i = blockIdx.x * blockDim.x + threadIdx.x;
  if (i >= TOT) return;
  unsigned* acc = (unsigned*)(ws + WS_ACC);
  int c = ((const int*)(ws + WS_CT))[i];
  bool keep = false;
  if (c > 0) keep = true;
  else if (c == 0) {
    unsigned key = ((const unsigned*)(ws + WS_KEY))[i];
    unsigned thr = acc[5];
    if (key > thr) keep = true;
    else if (key == thr) keep = atomicAdd((int*)acc + 4, 1) < ((int*)acc)[6];
  }
  if (keep) {
    int pos = atomicAdd((int*)acc + 3, 1);
    ((unsigned*)(ws + WS_LIST))[pos] = (unsigned)i;
  }
}

// ---- K6b: log-softmax CE over the dense kept list (wave per row, grid-stride) ----
__global__ void k_ce(const float* __restrict__ conf, char* ws) {
  const int wid = blockIdx.x * (blockDim.x >> 5) + (threadIdx.x >> 5);
  const int lane = threadIdx.x & 31;
  const int nw = gridDim.x * (blockDim.x >> 5);
  unsigned* acc = (unsigned*)(ws + WS_ACC);
  const int nk = ((const int*)acc)[3];
  const unsigned* list = (const unsigned*)(ws + WS_LIST);
  float local_ce = 0.f; int any = 0;
  for (int idx = wid; idx < nk; idx += nw) {
    int row = (int)list[idx];
    size_t base = (size_t)row * NCLS;
    float v0 = conf[base + lane];
    float v1 = conf[base + 32 + lane];
    float v2 = (lane < NCLS - 64) ? conf[base + 64 + lane] : -INFINITY;
    float m = fmaxf(fmaxf(v0, v1), v2);
    #pragma unroll
    for (int off = 16; off; off >>= 1) m = fmaxf(m, __shfl_xor(m, off, 32));
    float s = __expf(v0 - m) + __expf(v1 - m) +
              ((lane < NCLS - 64) ? __expf(v2 - m) : 0.f);
    #pragma unroll
    for (int off = 16; off; off >>= 1) s += __shfl_xor(s, off, 32);
    int c = ((const int*)(ws + WS_CT))[row];
    int ctc = min(max(c, 0), NCLS - 1);
    local_ce += (m + logf(s)) - conf[base + ctc];
    any = 1;
  }
  if (lane == 0 && any) atomicAdd((float*)acc + 2, local_ce);
}

// ---- K7: finalize ----
__global__ void k_fin(const char* ws, float* out) {
  const unsigned* acc = (const unsigned*)(ws + WS_ACC);
  out[0] = ((const float*)acc)[0] / (float)max(((const int*)acc)[1], 1);
  out[1] = ((const float*)acc)[2] / (float)max(((const int*)acc)[3], 1);
}

extern "C" void kernel_launch(void* const* d_in, const int* in_sizes, int n_in,
                              void* d_out, int out_size, void* d_ws, size_t ws_size,
                              hipStream_t stream) {
  const float* loc    = (const float*)d_in[0];
  const float* conf   = (const float*)d_in[1];
  const float* priors = (const float*)d_in[2];
  const float* gt     = (const float*)d_in[3];
  const int*   glab   = (const int*)d_in[4];
  char* ws = (char*)d_ws;
  float* out = (float*)d_out;

  const int zeroWords = WS_BTO / 4;
  k_zero <<<(zeroWords + 255) / 256, 256, 0, stream>>>((unsigned*)ws, zeroWords);
  k_match<<<dim3((NPRI + 255) / 256, BATCH), 256, 0, stream>>>(
      (const float4*)priors, (const float4*)gt, ws);
  k_force<<<1, 32, 0, stream>>>(ws);
  k_loc  <<<(TOT / 2 + 255) / 256, 256, 0, stream>>>(
      (const float4*)loc, (const float4*)priors, (const float4*)gt, glab, ws);
  k_score<<<TOT / 64, 256, 0, stream>>>(conf, ws);           // 8 rows/wave
  k_histA<<<(TOT + 255) / 256, 256, 0, stream>>>(ws);
  k_scanA<<<1, 1, 0, stream>>>(ws);
  k_histB<<<(TOT + 255) / 256, 256, 0, stream>>>(ws);
  k_scanB<<<1, 1, 0, stream>>>(ws);
  k_histC<<<(TOT + 255) / 256, 256, 0, stream>>>(ws);
  k_scanC<<<1, 1, 0, stream>>>(ws);
  k_select<<<(TOT + 255) / 256, 256, 0, stream>>>(ws);
  k_ce   <<<512, 256, 0, stream>>>(conf, ws);
  k_fin  <<<1, 1, 0, stream>>>(ws, out);
}